// CenteredBilinearPooling2_88373247082949
// MI455X (gfx1250) — compile-verified
//
#include <hip/hip_runtime.h>

typedef float v2f __attribute__((ext_vector_type(2)));
typedef float v8f __attribute__((ext_vector_type(8)));

#define LW      15
#define TN      4096
#define CC      64
#define ICNT    2048          // TN / STRIDE
#define ROWLEN  4160          // 64 + 64*64
#define WAVES   8
#define LSTRIDE 68            // 64 + 4 pad floats to de-phase LDS banks per row

__global__ __launch_bounds__(256)
void cbp2_kernel(const float* __restrict__ x,
                 const float* __restrict__ w1,
                 const float* __restrict__ w2,
                 float* __restrict__ out)
{
    __shared__ float winLDS[WAVES][16][LSTRIDE];

    const int tid  = threadIdx.x;
    const int wave = tid >> 5;
    const int lane = tid & 31;
    const int gw   = blockIdx.x * WAVES + wave;   // 0 .. 16383
    const int b    = gw / ICNT;
    const int i    = gw % ICNT;

    const float* xb = x + (size_t)b * TN * CC;

    // Uniform weights -> scalar regs
    float w1r[LW];
    float w2p[16];
#pragma unroll
    for (int k = 0; k < LW; ++k) { w1r[k] = w1[k]; w2p[k] = w2[k]; }
    w2p[15] = 0.0f;                                // K padding

    // ---- Stage 1: build centered window rows (xc = x - mu) into LDS ----
    const int c0 = lane;            // this lane's two channels
    const int c1 = lane + 32;
    float mu7a = 0.0f, mu7b = 0.0f; // mu at t = 2i (window row l == 7)

#pragma unroll 1
    for (int l = 0; l < LW; ++l) {
        const int t = 2 * i + l - 7;
        float va = 0.0f, vb = 0.0f;
        if (t >= 0 && t < TN) {
            float mua = 0.0f, mub = 0.0f;
#pragma unroll
            for (int k = 0; k < LW; ++k) {
                const int tt = t + k - 7;
                if (tt >= 0 && tt < TN) {
                    mua = fmaf(w1r[k], xb[(size_t)tt * CC + c0], mua);
                    mub = fmaf(w1r[k], xb[(size_t)tt * CC + c1], mub);
                }
            }
            va = xb[(size_t)t * CC + c0] - mua;
            vb = xb[(size_t)t * CC + c1] - mub;
            if (l == 7) { mu7a = mua; mu7b = mub; }
        }
        winLDS[wave][l][c0] = va;
        winLDS[wave][l][c1] = vb;
    }
    winLDS[wave][15][c0] = 0.0f;    // zero K-pad row
    winLDS[wave][15][c1] = 0.0f;

    __syncthreads();

    // ---- Stage 2: sigma = winT * diag(w2) * win via V_WMMA_F32_16X16X4_F32 ----
    const int half = lane >> 4;     // 0: lanes 0-15, 1: lanes 16-31
    const int j    = lane & 15;

    v8f acc[16];                    // 4x4 grid of 16x16 f32 tiles, kept in VGPRs
#pragma unroll
    for (int t16 = 0; t16 < 16; ++t16)
#pragma unroll
        for (int e = 0; e < 8; ++e) acc[t16][e] = 0.0f;

#pragma unroll 1
    for (int kk = 0; kk < 4; ++kk) {            // K-chunks of 4 (K = 16)
        const int l0 = 4 * kk + 2 * half;       // per-lane K rows {l0, l0+1}
        float r0[4], r1[4];                      // win[l0][16q+j], win[l0+1][16q+j]
#pragma unroll
        for (int q = 0; q < 4; ++q) {
            r0[q] = winLDS[wave][l0    ][16 * q + j];
            r1[q] = winLDS[wave][l0 + 1][16 * q + j];
        }
        const float s0 = w2p[l0], s1 = w2p[l0 + 1];
#pragma unroll
        for (int ci = 0; ci < 4; ++ci) {
            v2f a; a.x = s0 * r0[ci]; a.y = s1 * r1[ci];   // A = diag(w2)*win, col block ci
#pragma unroll
            for (int di = 0; di < 4; ++di) {
                v2f bb; bb.x = r0[di]; bb.y = r1[di];      // B = win, col block di
                acc[ci * 4 + di] = __builtin_amdgcn_wmma_f32_16x16x4_f32(
                    false, a, false, bb, (short)0, acc[ci * 4 + di], false, false);
            }
        }
    }

    // ---- Stage 3: L2 norm over the whole 4160-row, then scale + store ----
    float ssq = mu7a * mu7a + mu7b * mu7b;
#pragma unroll
    for (int t16 = 0; t16 < 16; ++t16)
#pragma unroll
        for (int e = 0; e < 8; ++e) ssq = fmaf(acc[t16][e], acc[t16][e], ssq);

#pragma unroll
    for (int off = 16; off >= 1; off >>= 1)
        ssq += __shfl_xor(ssq, off, 32);        // wave32 butterfly reduction

    const float denom = rsqrtf(fmaxf(ssq, 1e-12f));

    float* orow = out + (size_t)gw * ROWLEN;
    orow[c0] = mu7a * denom;
    orow[c1] = mu7b * denom;

#pragma unroll
    for (int ci = 0; ci < 4; ++ci)
#pragma unroll
        for (int di = 0; di < 4; ++di) {
            const v8f a8 = acc[ci * 4 + di];
#pragma unroll
            for (int e = 0; e < 8; ++e) {
                const int m = e + 8 * half;     // D tile layout: VGPR e -> M = e (+8 upper half)
                orow[64 + (16 * ci + m) * 64 + (16 * di + j)] = a8[e] * denom;
            }
        }
}

extern "C" void kernel_launch(void* const* d_in, const int* in_sizes, int n_in,
                              void* d_out, int out_size, void* d_ws, size_t ws_size,
                              hipStream_t stream) {
    (void)in_sizes; (void)n_in; (void)d_ws; (void)ws_size; (void)out_size;
    const float* x  = (const float*)d_in[0];
    const float* w1 = (const float*)d_in[1];
    const float* w2 = (const float*)d_in[2];
    float* out = (float*)d_out;

    const int total_waves = 8 * ICNT;           // B * I = 16384
    dim3 grid(total_waves / WAVES);             // 2048 blocks
    dim3 block(256);                            // 8 waves (wave32)
    cbp2_kernel<<<grid, block, 0, stream>>>(x, w1, w2, out);
}